// L2SquaredConv2d_50139448213964
// MI455X (gfx1250) — compile-verified
//
#include <hip/hip_runtime.h>
#include <hip/hip_bf16.h>

typedef __attribute__((ext_vector_type(2))) float v2f;
typedef __attribute__((ext_vector_type(4))) float v4f;
typedef __attribute__((ext_vector_type(8))) float v8f;

#define BD  32
#define CD  512
#define HWD 784
#define PD  2000
#define MTOT (BD * HWD)   // 25088

// ---------------------------------------------------------------------------
// Prep 1: xx[m] = sum_c input[b,c,hw]^2   (m = b*784 + hw)
// ---------------------------------------------------------------------------
__global__ __launch_bounds__(256) void xx_kernel(const float* __restrict__ inp,
                                                 float* __restrict__ xx) {
  int m = blockIdx.x * blockDim.x + threadIdx.x;
  if (m >= MTOT) return;
  int b  = m / HWD;
  int hw = m - b * HWD;
  const float* p = inp + (size_t)b * CD * HWD + hw;
  float s = 0.f;
#pragma unroll 8
  for (int c = 0; c < CD; ++c) {
    float v = p[(size_t)c * HWD];
    s += v * v;
  }
  xx[m] = s;
}

// ---------------------------------------------------------------------------
// Prep 2: ww[p] = sum_c W[p,c]^2
// ---------------------------------------------------------------------------
__global__ __launch_bounds__(256) void ww_kernel(const float* __restrict__ wgt,
                                                 float* __restrict__ ww) {
  int pidx = blockIdx.x * blockDim.x + threadIdx.x;
  if (pidx >= PD) return;
  const v4f* p = (const v4f*)(wgt + (size_t)pidx * CD);
  float s = 0.f;
#pragma unroll 4
  for (int c = 0; c < CD / 4; ++c) {
    v4f v = p[c];
    s += v.x * v.x + v.y * v.y + v.z * v.z + v.w * v.w;
  }
  ww[pidx] = s;
}

// ---------------------------------------------------------------------------
// Main GEMM: D = A(25088x512) x B(512x2000) via V_WMMA_F32_16X16X4_F32.
// Wave tile: 64M x 32N = 4(M) x 2(N) WMMA acc tiles (64 acc VGPRs).
// Block: 8 waves stacked in M = 512M x 32N.
// Grid: (25088/512, ceil(2000/32)) = (49, 63); last N-block's upper 16
// columns are clamped on load and skipped (wave-uniformly) on store.
// Epilogue: out = relu(xx - 2D + ww), nontemporal 16B stores (keep L2 for
// the input/weights working set, stream the 200MB output past it).
// ---------------------------------------------------------------------------
__global__ __launch_bounds__(256) void gemm_l2_kernel(
    const float* __restrict__ inp, const float* __restrict__ wgt,
    const float* __restrict__ xx, const float* __restrict__ ww,
    float* __restrict__ out) {
  const int lane = threadIdx.x & 31;
  const int wave = threadIdx.x >> 5;
  const int half = lane >> 4;   // 0: lanes 0-15, 1: lanes 16-31
  const int mrow = lane & 15;

  const int m0 = blockIdx.x * 512 + wave * 64;  // wave's first M
  const int n0 = blockIdx.y * 32;               // wave's first N
  const int n1 = n0 + mrow;                     // column in N-tile 0 (always < PD)
  const int n2 = n0 + 16 + mrow;                // column in N-tile 1
  const int n2c = (n2 < PD) ? n2 : (PD - 1);    // clamped for loads
  const bool n2ok = (n0 + 16) < PD;             // wave-uniform store predicate

  // B fragment sources: per lane the contiguous pair (K=k+2h, K=k+2h+1)
  const float* b0Ptr = wgt + (size_t)n1 * CD + 2 * half;
  const float* b1Ptr = wgt + (size_t)n2c * CD + 2 * half;

  // A fragment sources for 4 M-tiles (lane = row mrow, K offset 2*half).
  // 16-aligned M tiles never straddle a batch boundary (784 = 49*16).
  const float* a0p;
  const float* a1p;
  const float* a2p;
  const float* a3p;
  {
    int m, b, hw;
    m = m0 + 0 * 16 + mrow; b = m / HWD; hw = m - b * HWD;
    a0p = inp + ((size_t)b * CD + 2 * half) * HWD + hw;
    m = m0 + 1 * 16 + mrow; b = m / HWD; hw = m - b * HWD;
    a1p = inp + ((size_t)b * CD + 2 * half) * HWD + hw;
    m = m0 + 2 * 16 + mrow; b = m / HWD; hw = m - b * HWD;
    a2p = inp + ((size_t)b * CD + 2 * half) * HWD + hw;
    m = m0 + 3 * 16 + mrow; b = m / HWD; hw = m - b * HWD;
    a3p = inp + ((size_t)b * CD + 2 * half) * HWD + hw;
  }

  v8f acc[4][2];
#pragma unroll
  for (int t = 0; t < 4; ++t) {
    acc[t][0] = (v8f){};
    acc[t][1] = (v8f){};
  }

#pragma unroll 2
  for (int k = 0; k < CD; k += 4) {
    v2f bf0 = *(const v2f*)(b0Ptr + k);
    v2f bf1 = *(const v2f*)(b1Ptr + k);

    v2f af[4];
    af[0].x = a0p[(size_t)k * HWD];     af[0].y = a0p[(size_t)(k + 1) * HWD];
    af[1].x = a1p[(size_t)k * HWD];     af[1].y = a1p[(size_t)(k + 1) * HWD];
    af[2].x = a2p[(size_t)k * HWD];     af[2].y = a2p[(size_t)(k + 1) * HWD];
    af[3].x = a3p[(size_t)k * HWD];     af[3].y = a3p[(size_t)(k + 1) * HWD];

#pragma unroll
    for (int t = 0; t < 4; ++t) {
      acc[t][0] = __builtin_amdgcn_wmma_f32_16x16x4_f32(
          false, af[t], false, bf0, (short)0, acc[t][0], false, false);
      acc[t][1] = __builtin_amdgcn_wmma_f32_16x16x4_f32(
          false, af[t], false, bf1, (short)0, acc[t][1], false, false);
    }
  }

  // Epilogue. D layout: VGPR r, lanes 0-15 -> (M=r, N=lane);
  // lanes 16-31 -> (M=8+r, N=lane-16): per lane, the 8 acc entries of a tile
  // are 8 consecutive m (= consecutive hw) at fixed n.
  const float ww1 = ww[n1];
  const float ww2 = ww[n2c];

#pragma unroll
  for (int t = 0; t < 4; ++t) {
    int mbase = m0 + t * 16 + half * 8;
    int b  = mbase / HWD;
    int hw = mbase - b * HWD;
    const float* xxp = xx + mbase;

    float xv[8];
#pragma unroll
    for (int i = 0; i < 8; ++i) xv[i] = xxp[i];

    // N-tile 0 (always valid)
    {
      float* op = out + ((size_t)b * PD + n1) * HWD + hw;
      float r[8];
#pragma unroll
      for (int i = 0; i < 8; ++i) {
        float v = xv[i] - 2.0f * acc[t][0][i] + ww1;
        r[i] = v > 0.f ? v : 0.f;
      }
      v4f lo = {r[0], r[1], r[2], r[3]};
      v4f hi = {r[4], r[5], r[6], r[7]};
      __builtin_nontemporal_store(lo, (v4f*)(op));
      __builtin_nontemporal_store(hi, (v4f*)(op + 4));
    }
    // N-tile 1 (skipped wave-uniformly in the last N block)
    if (n2ok) {
      float* op = out + ((size_t)b * PD + n2) * HWD + hw;
      float r[8];
#pragma unroll
      for (int i = 0; i < 8; ++i) {
        float v = xv[i] - 2.0f * acc[t][1][i] + ww2;
        r[i] = v > 0.f ? v : 0.f;
      }
      v4f lo = {r[0], r[1], r[2], r[3]};
      v4f hi = {r[4], r[5], r[6], r[7]};
      __builtin_nontemporal_store(lo, (v4f*)(op));
      __builtin_nontemporal_store(hi, (v4f*)(op + 4));
    }
  }
}

// ---------------------------------------------------------------------------
extern "C" void kernel_launch(void* const* d_in, const int* in_sizes, int n_in,
                              void* d_out, int out_size, void* d_ws,
                              size_t ws_size, hipStream_t stream) {
  const float* inp = (const float*)d_in[0];   // [32, 512, 28, 28]
  const float* wgt = (const float*)d_in[1];   // [2000, 512, 1, 1]
  float* out = (float*)d_out;                 // [32, 2000, 28, 28]

  float* xx = (float*)d_ws;                   // 25088 floats
  float* ww = xx + MTOT;                      // 2000 floats

  xx_kernel<<<(MTOT + 255) / 256, 256, 0, stream>>>(inp, xx);
  ww_kernel<<<(PD + 255) / 256, 256, 0, stream>>>(wgt, ww);

  dim3 grid(MTOT / 512, (PD + 31) / 32);      // (49, 63)
  gemm_l2_kernel<<<grid, 256, 0, stream>>>(inp, wgt, xx, ww, out);
}